// ChemSageBlock_89206470738295
// MI455X (gfx1250) — compile-verified
//
#include <hip/hip_runtime.h>

#define D 128
#define TILE 16
#define BN_EPS 1e-5f
#define LDS_STRIDE 132   // 128 + 4 pad: lane->bank stride 4, conflict-free for 16 lanes

typedef __attribute__((ext_vector_type(2))) float v2f;
typedef __attribute__((ext_vector_type(8))) float v8f;

// ---------------------------------------------------------------- zero scratch
__global__ void zero_kernel(float* __restrict__ p, long n) {
    long i = (long)blockIdx.x * blockDim.x + threadIdx.x;
    long stride = (long)gridDim.x * blockDim.x;
    for (; i < n; i += stride) p[i] = 0.0f;
}

// ------------------------------------------------- scatter-sum: one wave / edge
__global__ void scatter_kernel(const float* __restrict__ x,
                               const long long* __restrict__ ei,
                               float* __restrict__ agg, float* __restrict__ deg,
                               int E) {
    int gid  = blockIdx.x * blockDim.x + threadIdx.x;
    int e    = gid >> 5;        // wave32 per edge
    int lane = gid & 31;
    if (e >= E) return;
    long long s = ei[e];
    long long d = ei[(long long)E + e];
    const float4 v = *(const float4*)(x + s * D + lane * 4);   // 16B aligned
    float* a = agg + d * D + lane * 4;
    atomicAdd(a + 0, v.x);
    atomicAdd(a + 1, v.y);
    atomicAdd(a + 2, v.z);
    atomicAdd(a + 3, v.w);
    if (lane == 0) atomicAdd(deg + d, 1.0f);
}

// ------------- fused: mean = agg/deg ; h = relu(mean@Wl^T + b + x@Wr^T) ; stats
__global__ void __launch_bounds__(256)
fused_gemm_kernel(const float* __restrict__ x, const float* __restrict__ agg,
                  const float* __restrict__ deg,
                  const float* __restrict__ W_l, const float* __restrict__ b_l,
                  const float* __restrict__ W_r,
                  float* __restrict__ h,
                  float* __restrict__ colsum, float* __restrict__ colsum2) {
    __shared__ float smMean[TILE * LDS_STRIDE];
    __shared__ float smX[TILE * LDS_STRIDE];

    const int row0 = blockIdx.x * TILE;
    const int tid  = threadIdx.x;

    // Stage 16x128 tiles of x and mean into LDS (512 float4 each, 2 per thread).
    for (int i = tid; i < TILE * (D / 4); i += 256) {
        int r  = i >> 5;          // 32 float4 per row
        int c4 = i & 31;
        int gr = row0 + r;
        float4 xa = *(const float4*)(x   + (long)gr * D + c4 * 4);
        float4 ag = *(const float4*)(agg + (long)gr * D + c4 * 4);
        float inv = 1.0f / fmaxf(deg[gr], 1.0f);
        float4 mn = make_float4(ag.x * inv, ag.y * inv, ag.z * inv, ag.w * inv);
        *(float4*)(smX    + r * LDS_STRIDE + c4 * 4) = xa;
        *(float4*)(smMean + r * LDS_STRIDE + c4 * 4) = mn;
    }
    __syncthreads();

    const int wave = tid >> 5;
    const int lane = tid & 31;
    const int j0   = wave * TILE;      // this wave's output-column tile
    const int m    = lane & 15;        // A row / B,C column within tile
    const int hi   = lane >> 4;        // half-wave selector (K offset +2 / M offset +8)

    v8f c = {0.f, 0.f, 0.f, 0.f, 0.f, 0.f, 0.f, 0.f};

    // B(k, n) = W[n][k]  (W row-major [out][in]) -> read along W's row n = j0+m
    const float* wlRow = W_l + (long)(j0 + m) * D;
    const float* wrRow = W_r + (long)(j0 + m) * D;
    const float* aMp   = smMean + m * LDS_STRIDE;
    const float* aXp   = smX    + m * LDS_STRIDE;

    for (int k0 = 0; k0 < D; k0 += 4) {
        const int kk = k0 + hi * 2;
        v2f am; am.x = aMp[kk]; am.y = aMp[kk + 1];
        v2f ax; ax.x = aXp[kk]; ax.y = aXp[kk + 1];
        float2 bl = *(const float2*)(wlRow + kk);   // 8B aligned (kk even)
        float2 br = *(const float2*)(wrRow + kk);
        v2f vbl; vbl.x = bl.x; vbl.y = bl.y;
        v2f vbr; vbr.x = br.x; vbr.y = br.y;
        // full-precision fp32 WMMA: D = A*B + C  (16x16x4)
        c = __builtin_amdgcn_wmma_f32_16x16x4_f32(false, am, false, vbl,
                                                  (short)0, c, false, false);
        c = __builtin_amdgcn_wmma_f32_16x16x4_f32(false, ax, false, vbr,
                                                  (short)0, c, false, false);
    }

    // Epilogue: bias + ReLU, store h, accumulate per-column sum / sum-of-squares.
    const int   col  = j0 + m;
    const float bias = b_l[col];
    float s1 = 0.f, s2 = 0.f;
#pragma unroll
    for (int v = 0; v < 8; ++v) {
        float hv = fmaxf(c[v] + bias, 0.0f);
        s1 += hv;
        s2 += hv * hv;
        int row = row0 + v + hi * 8;            // C layout: vgpr v -> M=v (+8 hi half)
        h[(long)row * D + col] = hv;
    }
    atomicAdd(colsum  + col, s1);
    atomicAdd(colsum2 + col, s2);
}

// -------------------------------------------- batch-norm finalize (in place)
__global__ void bn_finalize_kernel(float* __restrict__ out,
                                   const float* __restrict__ colsum,
                                   const float* __restrict__ colsum2,
                                   const float* __restrict__ gamma,
                                   const float* __restrict__ beta,
                                   long total4, float invN) {
    long i = (long)blockIdx.x * blockDim.x + threadIdx.x;
    long stride = (long)gridDim.x * blockDim.x;
    for (; i < total4; i += stride) {
        int d0 = (int)((i * 4) & (D - 1));
        float4 hv = *(float4*)(out + i * 4);
        float r[4] = {hv.x, hv.y, hv.z, hv.w};
        float o[4];
#pragma unroll
        for (int k = 0; k < 4; ++k) {
            int   dc  = d0 + k;
            float mu  = colsum[dc] * invN;
            float var = colsum2[dc] * invN - mu * mu;
            float sc  = gamma[dc] * rsqrtf(var + BN_EPS);
            o[k] = (r[k] - mu) * sc + beta[dc];
        }
        *(float4*)(out + i * 4) = make_float4(o[0], o[1], o[2], o[3]);
    }
}

// ---------------------------------------------------------------------- launch
extern "C" void kernel_launch(void* const* d_in, const int* in_sizes, int n_in,
                              void* d_out, int out_size, void* d_ws, size_t ws_size,
                              hipStream_t stream) {
    const float*     x     = (const float*)d_in[0];
    const long long* ei    = (const long long*)d_in[1];   // int64 edge_index [2, E]
    const float*     W_l   = (const float*)d_in[2];
    const float*     b_l   = (const float*)d_in[3];
    const float*     W_r   = (const float*)d_in[4];
    const float*     gamma = (const float*)d_in[5];
    const float*     beta  = (const float*)d_in[6];
    float*           out   = (float*)d_out;

    const int N = in_sizes[0] / D;      // 100000 (multiple of 16)
    const int E = in_sizes[1] / 2;      // 1600000

    // Workspace layout (floats): agg[N*D] | deg[N] | colsum[D] | colsum2[D]
    float* agg     = (float*)d_ws;
    float* deg     = agg + (long)N * D;
    float* colsum  = deg + N;
    float* colsum2 = colsum + D;

    const int threads = 256;

    long zeroN = (long)N * D + N + 2 * D;
    long zbL   = (zeroN + threads - 1) / threads;
    int  zb    = (int)(zbL < 4096 ? zbL : 4096);
    zero_kernel<<<zb, threads, 0, stream>>>(agg, zeroN);

    long scThreads = (long)E * 32;                        // one wave32 per edge
    int  sb = (int)((scThreads + threads - 1) / threads);
    scatter_kernel<<<sb, threads, 0, stream>>>(x, ei, agg, deg, E);

    int gb = N / TILE;                                    // 6250 row tiles
    fused_gemm_kernel<<<gb, 256, 0, stream>>>(x, agg, deg, W_l, b_l, W_r,
                                              out, colsum, colsum2);

    long total4 = (long)N * D / 4;
    long fbL    = (total4 + threads - 1) / threads;
    int  fb     = (int)(fbL < 8192 ? fbL : 8192);
    bn_finalize_kernel<<<fb, threads, 0, stream>>>(out, colsum, colsum2,
                                                   gamma, beta, total4,
                                                   1.0f / (float)N);
}